// Actor_DeepSet_38714835206233
// MI455X (gfx1250) — compile-verified
//
#include <hip/hip_runtime.h>
#include <hip/hip_bf16.h>

// MI455X / gfx1250, wave32. Fused DeepSet-actor MLP using V_WMMA_F32_16X16X4_F32
// (exact f32 math to match the f32 reference).
//
// Key algebra: state_other[b*8+a, p, j] == X[b*8+p+1, (j+a+1) & 63], so the
// deep-set branch is 8 GEMMs (one per feature-rotation s=a+1) against a single
// packed w1o, with rotation applied when building A-fragments from LDS.
// Round 2: w1o B-fragments register-resident across shifts; shifts processed in
// pairs (two independent accumulator chains sharing one B-frag) for XDL ILP;
// global_prefetch warming of the stage-2 weight slice.

typedef __attribute__((ext_vector_type(2))) float v2f;
typedef __attribute__((ext_vector_type(8))) float v8f;

#define OBS     64
#define HIDDEN  128
#define NUM_OUT 16

// d_ws layout (floats): pre-packed WMMA B-fragments.
// B-frag layout for 16x16x4 f32: element B(k,n) -> lane = n + 16*(k>>1), vgpr = k&1.
#define W1F_OFF   0        // [16 kt][8 nc][32 lane][2 v]  = 8192
#define W1OF_OFF  8192     // same shape                    = 8192
#define W2F_OFF   16384    // [64 kt][8 nc][32 lane][2 v]   = 32768
#define WVF_OFF   49152    // [32 kt][32 lane][2 v]         = 2048
#define WS_FLOATS 51200

__global__ __launch_bounds__(256) void pack_weights_kernel(
    const float* __restrict__ w1, const float* __restrict__ w1o,
    const float* __restrict__ w2, const float* __restrict__ wv,
    float* __restrict__ ws)
{
  int idx = blockIdx.x * 256 + threadIdx.x;
  if (idx >= WS_FLOATS) return;
  if (idx < W2F_OFF) {                       // w1 / w1o : [128n][64k], B(k,n)=W[n][k]
    int f = idx & 8191;
    const float* W = (idx < W1OF_OFF) ? w1 : w1o;
    int kt   = f >> 9;           // 0..15
    int rem  = f & 511;
    int nc   = rem >> 6;         // 0..7
    int rem2 = rem & 63;
    int lane = rem2 >> 1;
    int v    = rem2 & 1;
    int n = nc * 16 + (lane & 15);
    int k = kt * 4 + 2 * (lane >> 4) + v;
    ws[idx] = W[n * OBS + k];
  } else if (idx < WVF_OFF) {                // w2 : [128n][256k]
    int f = idx - W2F_OFF;
    int kt   = f >> 9;           // 0..63
    int rem  = f & 511;
    int nc   = rem >> 6;
    int rem2 = rem & 63;
    int lane = rem2 >> 1;
    int v    = rem2 & 1;
    int n = nc * 16 + (lane & 15);
    int k = kt * 4 + 2 * (lane >> 4) + v;
    ws[idx] = w2[n * (2 * HIDDEN) + k];
  } else {                                   // wv : [16n][128k]
    int f = idx - WVF_OFF;
    int kt  = f >> 6;            // 0..31
    int rem = f & 63;
    int lane = rem >> 1;
    int v    = rem & 1;
    int n = lane & 15;
    int k = kt * 4 + 2 * (lane >> 4) + v;
    ws[idx] = wv[n * HIDDEN + k];
  }
}

__device__ __forceinline__ v8f wmma_f32(v2f a, v2f b, v8f c) {
  // 8 args: (neg_a, A, neg_b, B, c_mod, C, reuse_a, reuse_b)
  return __builtin_amdgcn_wmma_f32_16x16x4_f32(false, a, false, b, (short)0, c,
                                               false, false);
}

__global__ __launch_bounds__(256) void actor_deepset_kernel(
    const float* __restrict__ X,
    const float* __restrict__ ws,
    const float* __restrict__ b1, const float* __restrict__ b1o,
    const float* __restrict__ b2, const float* __restrict__ bv,
    float* __restrict__ out)
{
  // LDS: X tile (stride 67 -> conflict-free A-frag reads since gcd(3,64)=1),
  // HF = h=[h1|hs] packed in stage-2 A-frag layout, H2F per-wave h2 A-frags,
  // S3P stage-3 split-K partials.  Total ~37 KB of the 320 KB WGP LDS.
  __shared__ __align__(16) float Xs[16 * 67];
  __shared__ __align__(16) float HF[64 * 32 * 2];     // 4096
  __shared__ __align__(16) float H2F[8 * 256];        // 2048
  __shared__ __align__(16) float S3P[8 * 256];        // 2048

  const int t    = threadIdx.x;
  const int w    = t >> 5;         // wave id = N-chunk (0..7)
  const int lane = t & 31;
  const int lh   = lane >> 4;      // half-wave (row-group selector in C/D)
  const int nlo  = lane & 15;
  const int r0   = blockIdx.x * 16;

  const float* W1F  = ws + W1F_OFF;
  const float* W1OF = ws + W1OF_OFF;
  const float* W2F  = ws + W2F_OFF;
  const float* WVF  = ws + WVF_OFF;

  const int nc = w;

  // Warm L2/WGP$ with this wave's stage-2 weight slice while stages A/B run.
  // (-> global_prefetch_b8; each (kt,nc) fragment chunk is 256 B.)
  __builtin_prefetch(W2F + (((size_t)lane * 8 + nc) * 32) * 2, 0, 3);
  __builtin_prefetch(W2F + (((size_t)(lane + 32) * 8 + nc) * 32) * 2, 0, 3);

  // ---- load 16x64 X tile (coalesced float4) ----
  {
    int m = t >> 4;                // element base e = t*4 -> row e>>6
    int j = (t & 15) * 4;
    const float4 xv = *(const float4*)(X + (size_t)(r0 + m) * OBS + j);
    float* dst = &Xs[m * 67 + j];
    dst[0] = xv.x; dst[1] = xv.y; dst[2] = xv.z; dst[3] = xv.w;
  }
  __syncthreads();

  const int c  = nc * 16 + nlo;    // this lane's h1-column within HIDDEN
  const float bias1  = b1[c];
  const float bias1o = b1o[c];
  const float bias2  = b2[c];

  // ---- stage A: h1 = relu(X @ w1^T + b1), 16 rows x 16 cols per wave ----
  {
    v8f d = {0.f, 0.f, 0.f, 0.f, 0.f, 0.f, 0.f, 0.f};
#pragma unroll
    for (int kt = 0; kt < 16; ++kt) {
      int kb = kt * 4 + 2 * lh;    // A(m,k): lane = m + 16*(k>>1), vgpr = k&1
      v2f a;
      a.x = Xs[nlo * 67 + kb];
      a.y = Xs[nlo * 67 + kb + 1];
      v2f b = *(const v2f*)(W1F + ((kt * 8 + nc) * 32 + lane) * 2);
      d = wmma_f32(a, b, d);
    }
#pragma unroll
    for (int dv = 0; dv < 8; ++dv) {       // D(m,n): m = dv + 8*lh, n = nlo
      float hv = fmaxf(d[dv] + bias1, 0.f);
      int m = dv + 8 * lh;
      HF[((c >> 2) * 32 + (m + 16 * ((c >> 1) & 1))) * 2 + (c & 1)] = hv;
    }
  }

  // ---- stage B: hs = (1/8) * sum_{q=1..7} relu(Xrot_s[group*8+q] @ w1o^T + b1o)
  //      shift s = agent+1; rotation applied while building A-frags.
  //      w1o B-frags live in registers (32 VGPRs) and are reused by all 8
  //      shifts; shifts run in pairs -> two independent WMMA chains per B-frag.
  {
    v2f bw[16];
#pragma unroll
    for (int kt = 0; kt < 16; ++kt)
      bw[kt] = *(const v2f*)(W1OF + ((kt * 8 + nc) * 32 + lane) * 2);

    float tacc[8];
    for (int sp = 0; sp < 4; ++sp) {
      const int s0 = 2 * sp + 1;
      const int s1 = 2 * sp + 2;
      v8f d0 = {0.f, 0.f, 0.f, 0.f, 0.f, 0.f, 0.f, 0.f};
      v8f d1 = {0.f, 0.f, 0.f, 0.f, 0.f, 0.f, 0.f, 0.f};
#pragma unroll
      for (int kt = 0; kt < 16; ++kt) {
        int kb = kt * 4 + 2 * lh;
        v2f a0, a1;
        a0.x = Xs[nlo * 67 + ((kb + s0) & 63)];
        a0.y = Xs[nlo * 67 + ((kb + 1 + s0) & 63)];
        a1.x = Xs[nlo * 67 + ((kb + s1) & 63)];
        a1.y = Xs[nlo * 67 + ((kb + 1 + s1) & 63)];
        d0 = wmma_f32(a0, bw[kt], d0);
        d1 = wmma_f32(a1, bw[kt], d1);
      }
      // per half-wave: rows m = dv (+8); within-group index q = dv. Sum q=1..7.
      float t0 = 0.f, t1 = 0.f;
#pragma unroll
      for (int dv = 1; dv < 8; ++dv) {
        t0 += fmaxf(d0[dv] + bias1o, 0.f);
        t1 += fmaxf(d1[dv] + bias1o, 0.f);
      }
      tacc[2 * sp]     = t0;
      tacc[2 * sp + 1] = t1;
    }
#pragma unroll
    for (int aa = 0; aa < 8; ++aa) {       // agent aa uses shift aa+1
      int m  = lh * 8 + aa;                // lanes 0-15: group 0, 16-31: group 1
      int c2 = 128 + c;
      HF[((c2 >> 2) * 32 + (m + 16 * ((c2 >> 1) & 1))) * 2 + (c2 & 1)] =
          tacc[aa] * 0.125f;
    }
  }
  __syncthreads();

  // ---- stage 2: h2 = relu(h @ w2^T + b2), K = 256 ----
  v8f d2 = {0.f, 0.f, 0.f, 0.f, 0.f, 0.f, 0.f, 0.f};
#pragma unroll 8
  for (int kt = 0; kt < 64; ++kt) {
    v2f a = *(const v2f*)&HF[(kt * 32 + lane) * 2];
    v2f b = *(const v2f*)(W2F + ((kt * 8 + nc) * 32 + lane) * 2);
    d2 = wmma_f32(a, b, d2);
  }

  // ---- stage 3: out = h2 @ wv^T + bv, split-K across waves.
  // Wave w produced h2 columns [w*16, w*16+16) == its own K-slice: re-pack
  // locally into A-frag layout, 4 WMMAs, then LDS reduce across waves.
  {
    float* myH2 = &H2F[w * 256];
#pragma unroll
    for (int dv = 0; dv < 8; ++dv) {
      float hv = fmaxf(d2[dv] + bias2, 0.f);
      int m = dv + 8 * lh;
      int n = nlo;                         // local k index
      myH2[((n >> 2) * 32 + (m + 16 * ((n >> 1) & 1))) * 2 + (n & 1)] = hv;
    }
    v8f d3 = {0.f, 0.f, 0.f, 0.f, 0.f, 0.f, 0.f, 0.f};
#pragma unroll
    for (int ktl = 0; ktl < 4; ++ktl) {
      v2f a = *(const v2f*)&myH2[(ktl * 32 + lane) * 2];
      v2f b = *(const v2f*)(WVF + ((w * 4 + ktl) * 32 + lane) * 2);
      d3 = wmma_f32(a, b, d3);
    }
#pragma unroll
    for (int dv = 0; dv < 8; ++dv) S3P[w * 256 + dv * 32 + lane] = d3[dv];
  }
  __syncthreads();

  // ---- cross-wave reduction + bias + store ----
  {
    int dv = t >> 5;
    int l2 = t & 31;
    float s = 0.f;
#pragma unroll
    for (int ww = 0; ww < 8; ++ww) s += S3P[ww * 256 + dv * 32 + l2];
    int m = dv + 8 * (l2 >> 4);
    int n = l2 & 15;
    out[(size_t)(r0 + m) * NUM_OUT + n] = s + bv[n];
  }
}

extern "C" void kernel_launch(void* const* d_in, const int* in_sizes, int n_in,
                              void* d_out, int out_size, void* d_ws, size_t ws_size,
                              hipStream_t stream) {
  const float* X   = (const float*)d_in[0];
  const float* w1  = (const float*)d_in[1];
  const float* b1  = (const float*)d_in[2];
  const float* w1o = (const float*)d_in[3];
  const float* b1o = (const float*)d_in[4];
  const float* w2  = (const float*)d_in[5];
  const float* b2  = (const float*)d_in[6];
  const float* wv  = (const float*)d_in[7];
  const float* bv  = (const float*)d_in[8];
  float* out = (float*)d_out;
  float* ws  = (float*)d_ws;

  const int R = in_sizes[0] / OBS;   // 131072 rows

  pack_weights_kernel<<<(WS_FLOATS + 255) / 256, 256, 0, stream>>>(
      w1, w1o, w2, wv, ws);
  actor_deepset_kernel<<<R / 16, 256, 0, stream>>>(
      X, ws, b1, b1o, b2, bv, out);
}